// RWKVRegionCell_27650999452038
// MI455X (gfx1250) — compile-verified
//
#include <hip/hip_runtime.h>
#include <hip/hip_bf16.h>

#define D 4096
#define KWTA_K 512
#define M_TIME 16

typedef __attribute__((ext_vector_type(2))) float v2f;
typedef __attribute__((ext_vector_type(8))) float v8f;

// ---------------------------------------------------------------------------
// Kernel 1: RMS norm.  x = x_in * rsqrt(mean(x_in^2) + 1e-6) * norm_w
// ---------------------------------------------------------------------------
__global__ __launch_bounds__(256) void rmsnorm_kernel(const float* __restrict__ x_in,
                                                      const float* __restrict__ norm_w,
                                                      float* __restrict__ xhat) {
    __shared__ float red[256];
    const int tid = threadIdx.x;
    float ssq = 0.0f;
    for (int i = tid; i < D; i += 256) {
        float t = x_in[i];
        ssq += t * t;
    }
    red[tid] = ssq;
    __syncthreads();
    for (int s = 128; s > 0; s >>= 1) {
        if (tid < s) red[tid] += red[tid + s];
        __syncthreads();
    }
    const float inv = rsqrtf(red[0] * (1.0f / (float)D) + 1e-6f);
    for (int i = tid; i < D; i += 256) {
        xhat[i] = x_in[i] * inv * norm_w[i];
    }
}

// ---------------------------------------------------------------------------
// Kernel 2: WMMA fp32 partial matvec over up to 3 matrices with K split into
// nseg segments.  wave w -> (mat, group, seg); computes 16 rows' partial dots
// over K range [seg*D/nseg, (seg+1)*D/nseg) and writes them to
//   partial[(mat*nseg + seg)*D + row].
// A-tile feed: one global_load_b64 per lane per WMMA (lane<16: K=k,k+1 of its
// row; lane>=16: K=k+2,k+3) -> contiguous row streaming, non-temporal.
// B: 4x16 with column 0 = x[k..k+3]; built with un-predicated LDS broadcast
// loads times a 0/1 lane mask (avoids exec-predicated ds_loads).
// ---------------------------------------------------------------------------
__global__ __launch_bounds__(256) void matvec_wmma_kernel(const float* __restrict__ W0,
                                                          const float* __restrict__ W1,
                                                          const float* __restrict__ W2,
                                                          const float* __restrict__ xin,
                                                          float* __restrict__ partial,
                                                          int nseg) {
    __shared__ float xs[D];
    for (int i = threadIdx.x; i < D; i += blockDim.x) xs[i] = xin[i];
    __syncthreads();

    const int lane = threadIdx.x & 31;
    const int wave = threadIdx.x >> 5;
    const int w    = blockIdx.x * (blockDim.x >> 5) + wave;

    const int seg  = w % nseg;
    const int grp  = (w / nseg) & 255;        // 256 row-groups of 16
    const int mat  = w / (nseg * 256);

    const float* W = (mat == 0) ? W0 : (mat == 1) ? W1 : W2;
    float* pout    = partial + ((size_t)(mat * nseg + seg)) * D;

    const int kseg = D / nseg;
    const int kbeg = seg * kseg;
    const int kend = kbeg + kseg;

    const int rbase = grp * 16;
    const int row   = rbase + (lane & 15);
    const int khalf = lane >> 4;                       // 0: lanes 0-15, 1: 16-31
    const float bm  = ((lane & 15) == 0) ? 1.0f : 0.0f; // B nonzero only lanes 0,16
    const float* wrow = W + (size_t)row * D;

    v8f acc = {};
    for (int k0 = kbeg; k0 < kend; k0 += 16) {
        const v2f wa0 = __builtin_nontemporal_load((const v2f*)(wrow + k0 +  0 + khalf * 2));
        const v2f wa1 = __builtin_nontemporal_load((const v2f*)(wrow + k0 +  4 + khalf * 2));
        const v2f wa2 = __builtin_nontemporal_load((const v2f*)(wrow + k0 +  8 + khalf * 2));
        const v2f wa3 = __builtin_nontemporal_load((const v2f*)(wrow + k0 + 12 + khalf * 2));
        const int xb = k0 + khalf;

#pragma unroll
        for (int u = 0; u < 4; ++u) {
            v2f A = (u == 0) ? wa0 : (u == 1) ? wa1 : (u == 2) ? wa2 : wa3;
            // B vgpr0: {lane0 -> x[kk], lane16 -> x[kk+1]},
            //   vgpr1: {lane0 -> x[kk+2], lane16 -> x[kk+3]}  (kk = k0+4u)
            v2f B;
            B.x = xs[xb + 4 * u] * bm;          // broadcast LDS read, 0/1 mask
            B.y = xs[xb + 4 * u + 2] * bm;
            acc = __builtin_amdgcn_wmma_f32_16x16x4_f32(
                /*neg_a=*/false, A, /*neg_b=*/false, B,
                /*c_mod=*/(short)0, acc, /*reuse_a=*/false, /*reuse_b=*/false);
        }
    }

    // D column 0: lane 0 holds M=0..7 in acc[0..7]; lane 16 holds M=8..15.
    if ((lane & 15) == 0) {
        const int mbase = rbase + khalf * 8;
#pragma unroll
        for (int j = 0; j < 8; ++j) {
            pout[mbase + j] = acc[j];
        }
    }
}

// ---------------------------------------------------------------------------
// Kernel 3: reduce r/k/v partials over segments + elementwise gate:
//   y = sigmoid(r) * ((num*lam + w*v) / (den*lam + w + eps)),
// rotary rotation on the first 2*M_TIME elements of v.
// ---------------------------------------------------------------------------
__global__ __launch_bounds__(256) void elementwise_kernel(const float* __restrict__ prr,
                                                          const float* __restrict__ prk,
                                                          const float* __restrict__ prv,
                                                          const float* __restrict__ state_num,
                                                          const float* __restrict__ state_den,
                                                          const float* __restrict__ decay,
                                                          const float* __restrict__ w_time,
                                                          const float* __restrict__ step_pos,
                                                          float* __restrict__ y,
                                                          int nseg) {
    const float sp = step_pos[0];
    for (int i = threadIdx.x + blockIdx.x * blockDim.x; i < D; i += gridDim.x * blockDim.x) {
        float rr = 0.0f, kk = 0.0f;
        for (int s = 0; s < nseg; ++s) {
            rr += prr[s * D + i];
            kk += prk[s * D + i];
        }
        float v;
        if (i < 2 * M_TIME) {
            const int p = i >> 1;
            float v0 = 0.0f, v1 = 0.0f;
            for (int s = 0; s < nseg; ++s) {
                v0 += prv[s * D + 2 * p];
                v1 += prv[s * D + 2 * p + 1];
            }
            const float th = sp * w_time[p];
            const float c = cosf(th);
            const float sn = sinf(th);
            v = (i & 1) ? (v0 * sn + v1 * c) : (v0 * c - v1 * sn);
        } else {
            v = 0.0f;
            for (int s = 0; s < nseg; ++s) v += prv[s * D + i];
        }
        const float r  = 1.0f / (1.0f + expf(-rr));
        const float w  = expf(kk);
        const float sd = 1.0f / (1.0f + expf(-decay[i]));
        const float lam = sd * sd;
        const float nn = state_num[i] * lam + w * v;
        const float nd = state_den[i] * lam + w;
        y[i] = r * (nn / (nd + 1e-9f));
    }
}

// ---------------------------------------------------------------------------
// Kernel 4: reduce W_o partials, h = x + Wo@y, then k-winner-take-all:
// 32-step radix select for the 512th-largest, threshold-write out.
// ---------------------------------------------------------------------------
__global__ __launch_bounds__(1024) void kwta_kernel(const float* __restrict__ prh,
                                                    const float* __restrict__ xhat,
                                                    float* __restrict__ out,
                                                    int nseg) {
    __shared__ unsigned keys[D];
    __shared__ float hval[D];
    __shared__ int cnt;
    const int tid = threadIdx.x;

    for (int i = tid; i < D; i += 1024) {
        float h = xhat[i];
        for (int s = 0; s < nseg; ++s) h += prh[s * D + i];
        hval[i] = h;
        unsigned u = __float_as_uint(h);
        keys[i] = (u & 0x80000000u) ? ~u : (u | 0x80000000u); // monotonic key
    }
    __syncthreads();

    unsigned prefix = 0;
    int remaining = KWTA_K;
    for (int b = 31; b >= 0; --b) {
        if (tid == 0) cnt = 0;
        __syncthreads();
        const unsigned ph = (b == 31) ? 0u : (prefix >> (b + 1));
        int local = 0;
        for (int i = tid; i < D; i += 1024) {
            const unsigned k = keys[i];
            const unsigned kh = (b == 31) ? 0u : (k >> (b + 1));
            if (kh == ph && ((k >> b) & 1u)) ++local;
        }
        if (local) atomicAdd(&cnt, local);
        __syncthreads();
        const int c = cnt;
        if (c >= remaining) prefix |= (1u << b);
        else remaining -= c;
        __syncthreads();
    }

    for (int i = tid; i < D; i += 1024) {
        out[i] = (keys[i] >= prefix) ? hval[i] : 0.0f;
    }
}

// ---------------------------------------------------------------------------
// Launch: norm -> fused {Wr,Wk,Wv}@x partials (WMMA, K-split) -> gate/rotary
//         -> Wo@y partials (WMMA, K-split) -> reduce + kWTA
// ---------------------------------------------------------------------------
extern "C" void kernel_launch(void* const* d_in, const int* in_sizes, int n_in,
                              void* d_out, int out_size, void* d_ws, size_t ws_size,
                              hipStream_t stream) {
    (void)in_sizes; (void)n_in; (void)out_size;
    const float* x_in      = (const float*)d_in[0];
    const float* state_num = (const float*)d_in[1];
    const float* state_den = (const float*)d_in[2];
    const float* norm_w    = (const float*)d_in[3];
    const float* W_r       = (const float*)d_in[4];
    const float* W_k       = (const float*)d_in[5];
    const float* W_v       = (const float*)d_in[6];
    const float* W_o       = (const float*)d_in[7];
    const float* decay     = (const float*)d_in[8];
    const float* w_time    = (const float*)d_in[9];
    const float* step_pos  = (const float*)d_in[10];
    float* out = (float*)d_out;

    // Choose K-split factor from available workspace (deterministic for a
    // given ws_size).  Need (2 + 4*nseg) * D floats.
    int nseg = 8;
    while (nseg > 1 && (size_t)(2 + 4 * nseg) * D * sizeof(float) > ws_size) nseg >>= 1;

    float* ws   = (float*)d_ws;
    float* xhat = ws;                    // D
    float* yv   = ws + D;                // D
    float* prr  = ws + 2 * D;            // nseg*D
    float* prk  = prr + (size_t)nseg * D;
    float* prv  = prk + (size_t)nseg * D;
    float* prh  = prv + (size_t)nseg * D;

    rmsnorm_kernel<<<1, 256, 0, stream>>>(x_in, norm_w, xhat);

    // Fused r/k/v: 3 matrices x 256 groups x nseg segments, 8 waves/block.
    matvec_wmma_kernel<<<96 * nseg, 256, 0, stream>>>(W_r, W_k, W_v, xhat, prr, nseg);

    elementwise_kernel<<<16, 256, 0, stream>>>(prr, prk, prv, state_num, state_den,
                                               decay, w_time, step_pos, yv, nseg);

    // W_o @ y: 1 matrix x 256 groups x nseg segments.
    matvec_wmma_kernel<<<32 * nseg, 256, 0, stream>>>(W_o, W_o, W_o, yv, prh, nseg);

    kwta_kernel<<<1, 1024, 0, stream>>>(prh, xhat, out, nseg);
}